// G2C_31516470018789
// MI455X (gfx1250) — compile-verified
//
#include <hip/hip_runtime.h>
#include <math.h>

// Problem constants (from reference setup): B=32, N=256, K=3, 100 steps.
#define BN 32
#define NN 256

typedef __attribute__((ext_vector_type(2))) float v2f;
typedef __attribute__((ext_vector_type(8))) float v8f;

// ---------------- block reduction (256 threads, wave32-safe) ----------------
__device__ __forceinline__ float block_reduce_sum_256(float v, float* red, int tid) {
  red[tid] = v;
  __syncthreads();
  for (int s = 128; s > 0; s >>= 1) {
    if (tid < s) red[tid] = red[tid] + red[tid + s];
    __syncthreads();
  }
  float r = red[0];
  __syncthreads();
  return r;
}

// ---------------- phase 1a: per-row sums of D^2 and mask --------------------
__global__ __launch_bounds__(256) void k_row_sums(const float* __restrict__ D,
                                                  const float* __restrict__ mask,
                                                  float* __restrict__ rows2,
                                                  float* __restrict__ maskrow,
                                                  float* __restrict__ maskcol0) {
  __shared__ float red[256];
  const int bi = blockIdx.x;          // b*256 + i
  const int j = threadIdx.x;
  const size_t idx = (size_t)bi * NN + j;
  const float d = D[idx];
  const float m = mask[idx];
  const float s = block_reduce_sum_256(d * d, red, j);
  const float ms = block_reduce_sum_256(m, red, j);
  if (j == 0) {
    rows2[bi] = s;
    maskrow[bi] = ms;
    maskcol0[bi] = mask[(size_t)bi * NN];   // mask[b, i, 0]
  }
}

// ---------------- phase 1b: per-batch scalars -------------------------------
__global__ __launch_bounds__(256) void k_batch_sums(const float* __restrict__ rows2,
                                                    const float* __restrict__ maskrow,
                                                    const float* __restrict__ maskcol0,
                                                    float* __restrict__ tot2,
                                                    float* __restrict__ denomB,
                                                    float* __restrict__ NmolB) {
  __shared__ float red[256];
  const int b = blockIdx.x, i = threadIdx.x;
  const float t = block_reduce_sum_256(rows2[b * NN + i], red, i);
  const float dn = block_reduce_sum_256(maskrow[b * NN + i], red, i);
  const float nm = block_reduce_sum_256(maskcol0[b * NN + i], red, i);
  if (i == 0) { tot2[b] = t; denomB[b] = dn; NmolB[b] = nm; }
}

// ------- phase 1c: build G (Gram) and interleaved PQ = {P0,P1,Q0,Q1} --------
// P = 4 m W / denom ; Q = P * D. PQ layout: [b*N + i][j/2][4] so the descent
// kernel fetches both (P,Q) for a j-pair with one global_load_b128.
__global__ __launch_bounds__(256) void k_build_gpq(const float* __restrict__ D,
                                                   const float* __restrict__ W,
                                                   const float* __restrict__ mask,
                                                   const float* __restrict__ rows2,
                                                   const float* __restrict__ tot2,
                                                   const float* __restrict__ denomB,
                                                   const float* __restrict__ NmolB,
                                                   float* __restrict__ G,
                                                   float* __restrict__ PQ) {
  const int bi = blockIdx.x;
  const int b = bi >> 8;
  const int j = threadIdx.x;
  const size_t idx = (size_t)bi * NN + j;
  const float d = D[idx], ww = W[idx], m = mask[idx];
  const float ri = rows2[bi];
  const float rj = rows2[(b << 8) + j];      // D symmetric -> axis-1 sum == row sum
  const float Nm = NmolB[b], tot = tot2[b], dn = denomB[b];
  const float invN = 1.0f / Nm;
  const float d2 = d * d;
  G[idx] = m * -0.5f * (d2 - rj * invN - ri * invN + tot * invN * invN);
  const float p = 4.0f * m * ww / dn;
  const size_t base = (size_t)bi * (NN * 2) + (size_t)(j >> 1) * 4 + (j & 1);
  PQ[base] = p;           // {P[j0], P[j1], ...}
  PQ[base + 2] = p * d;   // {..., Q[j0], Q[j1]}
}

// ---------------- phase 2: deflated power iteration + x init ----------------
__global__ __launch_bounds__(256) void k_power_init(const float* __restrict__ G,
                                                    const float* __restrict__ u_init,
                                                    const float* __restrict__ noise,
                                                    float* __restrict__ Xout) {
  __shared__ float u[256];
  __shared__ float up[3][256];
  __shared__ float red[256];
  const int b = blockIdx.x, i = threadIdx.x;
  const float* Gb = G + (size_t)b * NN * NN;

  for (int k = 0; k < 3; ++k) {
    u[i] = u_init[((size_t)k * BN + b) * NN + i];
    __syncthreads();
    for (int it = 0; it < 10; ++it) {
      const float s = block_reduce_sum_256(u[i] * u[i], red, i);
      const float inv = 1.0f / fmaxf(sqrtf(s), 1e-3f);
      float d0 = 0.0f, d1 = 0.0f;
      if (k > 0) d0 = block_reduce_sum_256(up[0][i] * u[i], red, i) * inv;
      if (k > 1) d1 = block_reduce_sum_256(up[1][i] * u[i], red, i) * inv;
      // (G u)_i via symmetry: read G column-major for coalescing
      float mv = 0.0f;
      for (int jj = 0; jj < NN; ++jj)
        mv = fmaf(Gb[(size_t)jj * NN + i], u[jj], mv);
      float v = mv * inv;
      if (k > 0) v -= d0 * up[0][i];
      if (k > 1) v -= d1 * up[1][i];
      __syncthreads();
      u[i] = v;
      __syncthreads();
    }
    const float eig = block_reduce_sum_256(u[i] * u[i], red, i);
    const float sc = 1.0f / sqrtf(sqrtf(eig + 0.01f));   // (eig+0.01)^-0.25
    up[k][i] = u[i] * sc;
    __syncthreads();
  }
  const size_t base = ((size_t)b * NN + i) * 3;
  Xout[base + 0] = up[0][i] + noise[base + 0];
  Xout[base + 1] = up[1][i] + noise[base + 1];
  Xout[base + 2] = up[2][i] + noise[base + 2];
}

// ---------------- phase 3: one stress-majorization step (WMMA) --------------
// Per block: one batch b, 64 rows (4 waves x 16-row tiles).
// Each wave computes a 16x16 f32 tile of C @ [X | 1 | 0...] with
// V_WMMA_F32_16X16X4_F32, accumulating K in chunks of 4.
// Output cols 0..2 = (C X)_i, col 3 = s_i = sum_j c_ij.
__global__ __launch_bounds__(128) void k_nlsq_step(const float* __restrict__ PQ,
                                                   const float* __restrict__ Xsrc,
                                                   float* __restrict__ Xdst,
                                                   const float* __restrict__ epsP,
                                                   const float* __restrict__ alphaP,
                                                   const float* __restrict__ alphaBaseP,
                                                   const float* __restrict__ TP,
                                                   float t) {
  __shared__ float Xs[256][4];          // batch coordinates (padded)
  __shared__ float stage[4][16][4];     // per-wave tile extraction: [wave][row][cx0..2,s]

  const int tid = threadIdx.x;
  const int b = blockIdx.x >> 2;        // 4 blocks per batch
  const int bt = blockIdx.x & 3;
  const int rowBase = bt * 64;

  const float* xsrc = Xsrc + (size_t)b * NN * 3;
  for (int idx = tid; idx < NN * 3; idx += 128)
    Xs[idx / 3][idx % 3] = xsrc[idx];
  __syncthreads();

  const int wave = tid >> 5;
  const int lane = tid & 31;
  const int half = lane >> 4;           // 0: K slots 0,1   1: K slots 2,3
  const int ln = lane & 15;
  const int i = rowBase + wave * 16 + ln;             // A-matrix row (M)
  const size_t rowPQ = ((size_t)b * NN + i) * (NN * 2);

  const float xi0 = Xs[i][0], xi1 = Xs[i][1], xi2 = Xs[i][2];

  // Hoisted, branchless B-fragment column selectors (lane-fixed):
  // B(k, N) = sel0*x + sel1*y + sel2*z + bias  -> cols 0..2 = X, col 3 = ones.
  const float sel0 = (ln == 0) ? 1.0f : 0.0f;
  const float sel1 = (ln == 1) ? 1.0f : 0.0f;
  const float sel2 = (ln == 2) ? 1.0f : 0.0f;
  const float bias = (ln == 3) ? 1.0f : 0.0f;

  v8f acc = {0.f, 0.f, 0.f, 0.f, 0.f, 0.f, 0.f, 0.f};

#pragma unroll 4
  for (int kb = 0; kb < NN / 4; ++kb) {
    const int j0 = kb * 4 + half * 2;   // this lane's two K columns
    const int j1 = j0 + 1;
    const float xa0 = Xs[j0][0], xa1 = Xs[j0][1], xa2 = Xs[j0][2];
    const float xb0 = Xs[j1][0], xb1 = Xs[j1][1], xb2 = Xs[j1][2];
    const float da0 = xi0 - xa0, da1 = xi1 - xa1, da2 = xi2 - xa2;
    const float db0 = xi0 - xb0, db1 = xi1 - xb1, db2 = xi2 - xb2;
    const float d2a = fmaf(da0, da0, fmaf(da1, da1, fmaf(da2, da2, 0.01f)));
    const float d2b = fmaf(db0, db0, fmaf(db1, db1, fmaf(db2, db2, 0.01f)));
    // one b128 load: {P[j0], P[j1], Q[j0], Q[j1]} (16B aligned)
    const float4 pq = *(const float4*)(PQ + rowPQ + (size_t)(2 * kb + half) * 4);
    // c_ij = P - Q * rsqrt(d2)  ( = 4 m W (DX - D) / (DX * denom) )
    const float ca = fmaf(-pq.z, rsqrtf(d2a), pq.x);
    const float cb = fmaf(-pq.w, rsqrtf(d2b), pq.y);
    v2f Av; Av.x = ca; Av.y = cb;       // 16x4 f32 A layout: v0=K0/K2, v1=K1/K3
    const float s0 = fmaf(sel0, xa0, fmaf(sel1, xa1, fmaf(sel2, xa2, bias)));
    const float s1 = fmaf(sel0, xb0, fmaf(sel1, xb1, fmaf(sel2, xb2, bias)));
    v2f Bv; Bv.x = s0; Bv.y = s1;
    acc = __builtin_amdgcn_wmma_f32_16x16x4_f32(false, Av, false, Bv,
                                                (short)0, acc, false, false);
  }

  // C/D 16x16 f32 layout: acc[r] holds (M = r + 8*half, N = ln)
  if (ln < 4) {
#pragma unroll
    for (int r = 0; r < 8; ++r) stage[wave][r + 8 * half][ln] = acc[r];
  }
  __syncthreads();

  if (tid < 64) {
    const int row = rowBase + tid;
    const int wv = tid >> 4, lr = tid & 15;
    const float cx0 = stage[wv][lr][0];
    const float cx1 = stage[wv][lr][1];
    const float cx2 = stage[wv][lr][2];
    const float s = stage[wv][lr][3];
    const float x0 = Xs[row][0], x1 = Xs[row][1], x2 = Xs[row][2];
    const float eps = epsP[0], alpha = alphaP[0], ab = alphaBaseP[0], T = TP[0];
    // g_i = s_i x_i - (C X)_i ;  dx = -eps g
    const float dx0 = -eps * (s * x0 - cx0);
    const float dx1 = -eps * (s * x1 - cx1);
    const float dx2 = -eps * (s * x2 - cx2);
    const float speed = sqrtf(fmaf(dx0, dx0, fmaf(dx1, dx1, fmaf(dx2, dx2, 0.001f))));
    const float alpha_t = ab + (alpha - ab) * ((T - t) / T);
    const float scale = alpha_t * tanhf(speed / alpha_t) / speed;
    float* o = Xdst + ((size_t)b * NN + row) * 3;
    o[0] = fmaf(dx0, scale, x0);
    o[1] = fmaf(dx1, scale, x1);
    o[2] = fmaf(dx2, scale, x2);
  }
}

// ---------------------------------------------------------------------------
extern "C" void kernel_launch(void* const* d_in, const int* in_sizes, int n_in,
                              void* d_out, int out_size, void* d_ws, size_t ws_size,
                              hipStream_t stream) {
  const float* D          = (const float*)d_in[0];
  const float* W          = (const float*)d_in[1];
  const float* mask       = (const float*)d_in[2];
  const float* u_init     = (const float*)d_in[3];
  const float* noise      = (const float*)d_in[4];
  const float* eps        = (const float*)d_in[5];
  const float* alpha      = (const float*)d_in[6];
  const float* alpha_base = (const float*)d_in[7];
  const float* T_param    = (const float*)d_in[8];
  float* out = (float*)d_out;

  float* w = (float*)d_ws;
  const size_t NG = (size_t)BN * NN * NN;   // 2,097,152 floats
  float* G        = w;                       // [BN*NN*NN]
  float* PQ       = w + NG;                  // [BN*NN*NN*2] interleaved P/Q
  float* rows2    = w + 3 * NG;
  float* maskrow  = rows2 + BN * NN;
  float* maskcol0 = maskrow + BN * NN;
  float* tot2     = maskcol0 + BN * NN;
  float* denomB   = tot2 + BN;
  float* NmolB    = denomB + BN;
  float* xPing    = NmolB + BN;             // [BN, NN, 3] ping buffer

  k_row_sums<<<BN * NN, 256, 0, stream>>>(D, mask, rows2, maskrow, maskcol0);
  k_batch_sums<<<BN, 256, 0, stream>>>(rows2, maskrow, maskcol0, tot2, denomB, NmolB);
  k_build_gpq<<<BN * NN, 256, 0, stream>>>(D, W, mask, rows2, tot2, denomB, NmolB, G, PQ);
  k_power_init<<<BN, 256, 0, stream>>>(G, u_init, noise, out);

  // 100 sequential descent steps, ping-pong so final step writes d_out.
  for (int t = 0; t < 100; ++t) {
    const float* src = (t & 1) ? xPing : out;
    float* dst = (t & 1) ? out : xPing;
    k_nlsq_step<<<BN * 4, 128, 0, stream>>>(PQ, src, dst, eps, alpha, alpha_base,
                                            T_param, (float)t);
  }
}